// EopPairCosineSimilarity_79723182949011
// MI455X (gfx1250) — compile-verified
//
#include <hip/hip_runtime.h>
#include <hip/hip_bf16.h>

// ---------------------------------------------------------------------------
// EOP-pair cosine similarity for MI455X (gfx1250, wave32).
//   out[0 .. B*S)       : cos(x_j, x_{(j+1)%cnt}) / TEMP for j < cnt, else -100
//   out[B*S .. 2*B*S)   : compacted labels (as float) for j < cnt, else -100
// Pass 1: per-row stable compaction (shared-memory scan) -> perm, cnt, labels.
// Pass 2: one wave per 16 pairs; C = A * B^T via V_WMMA_F32_16X16X4_F32,
//         diag(C) gives the 16 neighbor dot-products; norms via per-lane FMA
//         + one wave32 shfl_xor(16). Memory-bound (~67 MB valid traffic).
// ---------------------------------------------------------------------------

typedef float v2f __attribute__((ext_vector_type(2)));
typedef float v8f __attribute__((ext_vector_type(8)));

#define TEMP_INV 10.0f
#define PADV    -100.0f
#define EPSV     1e-8f

// ---------------------------------------------------------------------------
// Pass 1: stable compaction of valid (label != -100) positions per batch row.
// grid.x = B, blockDim.x = 256.
// ---------------------------------------------------------------------------
__global__ void eop_compact_kernel(const int* __restrict__ labels,
                                   float* __restrict__ out,
                                   int* __restrict__ perm,
                                   int* __restrict__ cnt,
                                   int B, int S) {
    const int b   = blockIdx.x;
    const int tid = threadIdx.x;
    const int T   = 256;

    __shared__ int sdata[256];
    __shared__ int sbase;

    float* cosrow = out + (size_t)b * S;
    float* labrow = out + (size_t)B * S + (size_t)b * S;

    // Fill both output halves of this row with the pad value.
    for (int s = tid; s < S; s += T) {
        cosrow[s] = PADV;
        labrow[s] = PADV;
    }
    if (tid == 0) sbase = 0;
    __syncthreads();

    // Chunked stable scan + scatter.
    for (int c0 = 0; c0 < S; c0 += T) {
        const int s   = c0 + tid;
        const int lab = labels[(size_t)b * S + s];
        const int m   = (lab != -100) ? 1 : 0;

        sdata[tid] = m;
        __syncthreads();
        // Inclusive Hillis-Steele scan over 256 elements.
        for (int off = 1; off < T; off <<= 1) {
            int v   = sdata[tid];
            int add = (tid >= off) ? sdata[tid - off] : 0;
            __syncthreads();
            sdata[tid] = v + add;
            __syncthreads();
        }
        const int incl = sdata[tid];
        const int base = sbase;
        if (m) {
            const int j = base + (incl - m);   // stable exclusive position
            perm[(size_t)b * S + j] = s;
            labrow[j] = (float)lab;
        }
        __syncthreads();
        if (tid == T - 1) sbase = base + incl;
        __syncthreads();
    }
    if (tid == 0) cnt[b] = sbase;
}

// ---------------------------------------------------------------------------
// Pass 2: one wave32 per tile of 16 pairs. C = A * B^T with f32 WMMA (K=4
// steps over D). A rows = x_{j0+m}; B rows = x_{(j0+m+1) % cnt}.
// grid = (S/16, B), blockDim.x = 32 (EXEC all ones for WMMA).
// ---------------------------------------------------------------------------
__global__ void eop_gram_wmma_kernel(const float* __restrict__ seq,
                                     const int* __restrict__ perm,
                                     const int* __restrict__ cnt_arr,
                                     float* __restrict__ out,
                                     int B, int S, int D) {
    const int b   = blockIdx.y;
    const int j0  = blockIdx.x << 4;          // 16 pairs per tile
    const int cnt = cnt_arr[b];
    if (j0 >= cnt) return;                    // uniform: whole wave exits

    const int lane = threadIdx.x;             // 0..31, wave32
    const int m    = lane & 15;               // matrix row (M / N) this lane feeds
    const int hi   = lane >> 4;               // 0: K%4 in {0,1}; 1: K%4 in {2,3}

    // Clamp padding lanes to safe rows; their results are never stored.
    const int j  = j0 + m;
    const int jc = (j < cnt) ? j : (cnt - 1);
    const int jn = (jc + 1 < cnt) ? (jc + 1) : 0;   // cyclic next
    const int rowA = perm[(size_t)b * S + jc];
    const int rowB = perm[(size_t)b * S + jn];

    const float* xa = seq + ((size_t)b * S + rowA) * D + 2 * hi;
    const float* xb = seq + ((size_t)b * S + rowB) * D + 2 * hi;

    v8f c = {};
    float na = 0.0f, nb = 0.0f;

    for (int k = 0; k < D; k += 4) {
        v2f a  = *(const v2f*)(xa + k);       // A frag: row m, cols k+2*hi+{0,1}
        v2f bb = *(const v2f*)(xb + k);       // B frag: col m, rows k+2*hi+{0,1}
        na += a.x * a.x + a.y * a.y;          // |x_j|^2 partial (this lane's Ks)
        nb += bb.x * bb.x + bb.y * bb.y;      // |x_{j+1}|^2 partial
        // C += A * B^T  (16x16x4 f32)
        c = __builtin_amdgcn_wmma_f32_16x16x4_f32(
                /*neg_a=*/false, a, /*neg_b=*/false, bb,
                /*c_mod=*/(short)0, c, /*reuse_a=*/false, /*reuse_b=*/false);
    }

    // Combine the two K-halves (lanes l and l^16 cover complementary Ks).
    na += __shfl_xor(na, 16, 32);
    nb += __shfl_xor(nb, 16, 32);

    // Extract diag(C): m<8 -> lane m, VGPR m;  m>=8 -> lane 16+m, VGPR m-8.
    float d = 0.0f;
    const int idx = (lane < 8) ? lane : ((lane >= 24) ? (lane - 24) : -1);
#pragma unroll
    for (int i = 0; i < 8; ++i)
        if (idx == i) d = c[i];
    const int src = (m < 8) ? m : (m + 16);
    d = __shfl(d, src, 32);                   // lanes 0..15 now hold dot(x_j, x_{j+1})

    if (hi == 0 && j < cnt) {
        const float nx = sqrtf(na);
        const float ny = sqrtf(nb);
        const float denom = fmaxf(nx * ny, EPSV);
        out[(size_t)b * S + j] = (d / denom) * TEMP_INV;
    }
}

// ---------------------------------------------------------------------------
// Host launcher.
// ---------------------------------------------------------------------------
extern "C" void kernel_launch(void* const* d_in, const int* in_sizes, int n_in,
                              void* d_out, int out_size, void* d_ws, size_t ws_size,
                              hipStream_t stream) {
    const float* seq    = (const float*)d_in[0];   // [B, S, D] f32
    const int*   labels = (const int*)d_in[1];     // [B, S] i32

    const int B = 8;
    const int S = in_sizes[1] / B;                 // 4096
    const int D = in_sizes[0] / in_sizes[1];       // 1024

    float* out = (float*)d_out;                    // [2, B, S] (cos, labels)

    int* perm = (int*)d_ws;                        // [B*S]
    int* cnt  = perm + (size_t)B * S;              // [B]

    eop_compact_kernel<<<B, 256, 0, stream>>>(labels, out, perm, cnt, B, S);

    dim3 grid(S / 16, B);
    eop_gram_wmma_kernel<<<grid, 32, 0, stream>>>(seq, perm, cnt, out, B, S, D);
}